// StationGNN_1331439862453
// MI455X (gfx1250) — compile-verified
//
#include <hip/hip_runtime.h>
#include <hip/hip_bf16.h>

// ---------------------------------------------------------------------------
// StationGNN on MI455X (gfx1250): GCNConv -> ReLU -> GATv2Conv + gated skip.
// Dense GEMMs use V_WMMA_F32_16X16X4_F32 (fp32-precise matrix pipe).
// Edge passes are wave-per-edge (wave32) with f32 atomics that resolve in the
// 192MB L2 (working set ~137MB).
// ---------------------------------------------------------------------------

typedef __attribute__((ext_vector_type(2))) float v2f;
typedef __attribute__((ext_vector_type(8))) float v8f;

#define NODE_IN 64
#define HID     64
#define EMB     32
#define HEADS   4
#define GATD    (HEADS * EMB)   // 128

// ------------------------- WMMA fp32 GEMM ----------------------------------
// C[nrows x OUT] = A[nrows x 64] * W[OUT x 64]^T (+ bias), K = 64.
// One wave per 16x16 output tile; 16 chained v_wmma_f32_16x16x4_f32 ops.
// A-layout (16x4 f32): lanes 0-15 hold M=lane {K=k,k+1}; lanes 16-31 {K=k+2,k+3}.
// B-layout (4x16 f32): mirrored with N in lanes.
// C/D layout: vgpr r -> M = r + (lane>=16 ? 8 : 0), N = lane&15.
template <int OUT>
__global__ void gemm_wmma_f32(const float* __restrict__ A,
                              const float* __restrict__ W,
                              const float* __restrict__ bias,
                              float* __restrict__ C, int nrows) {
    const int wave = (blockIdx.x * blockDim.x + threadIdx.x) >> 5;
    const int lane = threadIdx.x & 31;
    constexpr int colTiles = OUT / 16;
    const int mTile = wave / colTiles;
    const int oTile = wave % colTiles;
    if (mTile * 16 >= nrows) return;          // wave-uniform: EXEC stays all-1

    const int rowA = mTile * 16 + (lane & 15);
    const int colW = oTile * 16 + (lane & 15);
    const int kHi  = (lane >> 4) * 2;         // 0 for lanes 0-15, 2 for 16-31

    const float* __restrict__ aRow = A + (size_t)rowA * 64;
    const float* __restrict__ wRow = W + (size_t)colW * 64;

    v8f acc = {};
#pragma unroll
    for (int k = 0; k < 64; k += 4) {
        v2f a, b;
        a.x = aRow[k + kHi];
        a.y = aRow[k + kHi + 1];
        b.x = wRow[k + kHi];
        b.y = wRow[k + kHi + 1];
        acc = __builtin_amdgcn_wmma_f32_16x16x4_f32(
            /*neg_a=*/false, a, /*neg_b=*/false, b,
            /*c_mod=*/(short)0, acc, /*reuse_a=*/false, /*reuse_b=*/false);
    }

    const int col     = oTile * 16 + (lane & 15);
    const int rowBase = mTile * 16 + ((lane >> 4) ? 8 : 0);
    const float bc    = bias ? bias[col] : 0.0f;
#pragma unroll
    for (int r = 0; r < 8; ++r)
        C[(size_t)(rowBase + r) * OUT + col] = acc[r] + bc;
}

// ------------------------- helpers -----------------------------------------
__device__ __forceinline__ void atomicMaxF32(float* addr, float val) {
    if (val >= 0.0f)
        atomicMax((int*)addr, __float_as_int(val));
    else
        atomicMin((unsigned int*)addr, __float_as_uint(val));
}

// ------------------------- kernels -----------------------------------------
__global__ void init_ws_k(float* la_sum, float* cnt, float* agg,
                          float* denom, float* amax, float* outbuf, int Nn) {
    const int i = blockIdx.x * blockDim.x + threadIdx.x;
    if (i < Nn)       { la_sum[i] = 0.0f; cnt[i] = 0.0f; }
    if (i < Nn * 4)   { denom[i] = 0.0f; amax[i] = -__builtin_inff(); }
    if (i < Nn * 64)  { agg[i] = 0.0f; }
    if (i < Nn * 128) { outbuf[i] = 0.0f; }
}

__global__ void degree_k(const int* __restrict__ dst,
                         const float* __restrict__ w,
                         float* la_sum, float* cnt, int Ee) {
    const int e = blockIdx.x * blockDim.x + threadIdx.x;
    if (e >= Ee) return;
    const int d = dst[e];
    atomicAdd(&la_sum[d], w[e]);
    atomicAdd(&cnt[d], 1.0f);
}

// GCN neighbor aggregation: wave per edge, lane covers 2 of 64 channels.
// dinv = rsqrt(la_sum + 1) (self-loop weight 1 always makes deg > 0).
__global__ void gcn_scatter_k(const int* __restrict__ src,
                              const int* __restrict__ dst,
                              const float* __restrict__ w,
                              const float* __restrict__ la_sum,
                              const float* __restrict__ h,
                              float* __restrict__ agg, int Ee) {
    const int wave = (blockIdx.x * blockDim.x + threadIdx.x) >> 5;
    const int lane = threadIdx.x & 31;
    if (wave >= Ee) return;
    const int s = src[wave], d = dst[wave];
    const float norm = rsqrtf(la_sum[s] + 1.0f) * w[wave] * rsqrtf(la_sum[d] + 1.0f);
    const float* hs = h + (size_t)s * HID;
    float* ad = agg + (size_t)d * HID;
    atomicAdd(&ad[lane],      hs[lane]      * norm);
    atomicAdd(&ad[lane + 32], hs[lane + 32] * norm);
}

// hg = relu(agg + h * (1/deg) + b1)   (self-loop contribution h[i]*dinv^2 folded in)
__global__ void gcn_finalize_k(const float* __restrict__ agg,
                               const float* __restrict__ h,
                               const float* __restrict__ la_sum,
                               const float* __restrict__ b1,
                               float* __restrict__ hg, int Nn) {
    const int idx = blockIdx.x * blockDim.x + threadIdx.x;
    if (idx >= Nn * HID) return;
    const int i = idx >> 6, c = idx & 63;
    const float dinv2 = 1.0f / (la_sum[i] + 1.0f);
    const float v = agg[idx] + h[idx] * dinv2 + b1[c];
    hg[idx] = v > 0.0f ? v : 0.0f;
}

// alpha[e][h] = att[h] . leaky_relu(xl[s][h] + xr[d][h] + ea*We[h], 0.2); segment max.
__global__ void gat_alpha_k(const int* __restrict__ src,
                            const int* __restrict__ dst,
                            const float* __restrict__ w,
                            const float* __restrict__ la_sum,
                            const float* __restrict__ cnt,
                            const float* __restrict__ xl,
                            const float* __restrict__ xr,
                            const float* __restrict__ We,
                            const float* __restrict__ att,
                            float* __restrict__ alpha,
                            float* __restrict__ amax, int Ee, int Nn) {
    const int idx = blockIdx.x * blockDim.x + threadIdx.x;
    const int e = idx >> 2, hh = idx & 3;
    if (e >= Ee + Nn) return;
    int s, d; float ea;
    if (e < Ee) { s = src[e]; d = dst[e]; ea = w[e]; }
    else {                                  // self-loop, edge_attr = mean of incoming
        const int i = e - Ee; s = i; d = i;
        const float c = cnt[i];
        ea = c > 0.0f ? la_sum[i] / c : 0.0f;
    }
    const float* xls  = xl  + (size_t)s * GATD + hh * EMB;
    const float* xrd  = xr  + (size_t)d * GATD + hh * EMB;
    const float* Weh  = We  + hh * EMB;
    const float* atth = att + hh * EMB;
    float acc = 0.0f;
#pragma unroll
    for (int c = 0; c < EMB; ++c) {
        float m = xls[c] + xrd[c] + ea * Weh[c];
        m = m > 0.0f ? m : 0.2f * m;
        acc += atth[c] * m;
    }
    alpha[(size_t)e * 4 + hh] = acc;
    atomicMaxF32(&amax[(size_t)d * 4 + hh], acc);
}

__global__ void gat_expsum_k(const int* __restrict__ dst,
                             float* __restrict__ alpha,
                             const float* __restrict__ amax,
                             float* __restrict__ denom, int Ee, int Nn) {
    const int idx = blockIdx.x * blockDim.x + threadIdx.x;
    const int e = idx >> 2, hh = idx & 3;
    if (e >= Ee + Nn) return;
    const int d = (e < Ee) ? dst[e] : (e - Ee);
    const float ex = __expf(alpha[(size_t)e * 4 + hh] - amax[(size_t)d * 4 + hh]);
    alpha[(size_t)e * 4 + hh] = ex;
    atomicAdd(&denom[(size_t)d * 4 + hh], ex);
}

// out[d][h][:] += xl[s][h][:] * (ex / denom[d][h]); wave per edge, lane = channel.
__global__ void gat_scatter_k(const int* __restrict__ src,
                              const int* __restrict__ dst,
                              const float* __restrict__ xl,
                              const float* __restrict__ alpha,
                              const float* __restrict__ denom,
                              float* __restrict__ outbuf, int Ee, int Nn) {
    const int wave = (blockIdx.x * blockDim.x + threadIdx.x) >> 5;
    const int lane = threadIdx.x & 31;
    if (wave >= Ee + Nn) return;
    const int s = (wave < Ee) ? src[wave] : (wave - Ee);
    const int d = (wave < Ee) ? dst[wave] : (wave - Ee);
    const float* xls = xl + (size_t)s * GATD;
    float* od = outbuf + (size_t)d * GATD;
#pragma unroll
    for (int hh = 0; hh < HEADS; ++hh) {
        const float a = alpha[(size_t)wave * 4 + hh] /
                        (denom[(size_t)d * 4 + hh] + 1e-16f);
        atomicAdd(&od[hh * EMB + lane], xls[hh * EMB + lane] * a);
    }
}

// out = head_mean(outbuf) + bias_gat + skip_gate * skip
__global__ void final_out_k(const float* __restrict__ outbuf,
                            const float* __restrict__ skip,
                            const float* __restrict__ bias_gat,
                            const float* __restrict__ skip_gate,
                            float* __restrict__ out, int Nn) {
    const int idx = blockIdx.x * blockDim.x + threadIdx.x;
    if (idx >= Nn * EMB) return;
    const int i = idx >> 5, c = idx & 31;
    const float* ob = outbuf + (size_t)i * GATD;
    const float m = 0.25f * (ob[c] + ob[EMB + c] + ob[2 * EMB + c] + ob[3 * EMB + c]);
    out[idx] = m + bias_gat[c] + skip_gate[0] * skip[idx];
}

// ------------------------- launch ------------------------------------------
extern "C" void kernel_launch(void* const* d_in, const int* in_sizes, int n_in,
                              void* d_out, int out_size, void* d_ws, size_t ws_size,
                              hipStream_t stream) {
    const float* x   = (const float*)d_in[0];
    const int*   ei  = (const int*)  d_in[1];
    const float* w   = (const float*)d_in[2];
    const float* W1  = (const float*)d_in[3];
    const float* b1  = (const float*)d_in[4];
    const float* Wl  = (const float*)d_in[5];
    const float* bl  = (const float*)d_in[6];
    const float* Wr  = (const float*)d_in[7];
    const float* br  = (const float*)d_in[8];
    const float* We  = (const float*)d_in[9];
    const float* att = (const float*)d_in[10];
    const float* bg  = (const float*)d_in[11];
    const float* Ws  = (const float*)d_in[12];
    const float* bs  = (const float*)d_in[13];
    const float* sg  = (const float*)d_in[14];
    float* out = (float*)d_out;

    const int Nn = in_sizes[0] / NODE_IN;   // 50000
    const int Ee = in_sizes[2];             // 800000
    const int E2 = Ee + Nn;
    const int* src = ei;
    const int* dst = ei + Ee;

    // workspace layout (floats)
    float* ws = (float*)d_ws;
    size_t off = 0;
    float* la_sum = ws + off; off += (size_t)Nn;
    float* cnt    = ws + off; off += (size_t)Nn;
    float* h      = ws + off; off += (size_t)Nn * HID;
    float* agg    = ws + off; off += (size_t)Nn * HID;
    float* hg     = ws + off; off += (size_t)Nn * HID;
    float* xl     = ws + off; off += (size_t)Nn * GATD;
    float* xr     = ws + off; off += (size_t)Nn * GATD;
    float* amax   = ws + off; off += (size_t)Nn * HEADS;
    float* denom  = ws + off; off += (size_t)Nn * HEADS;
    float* outbuf = ws + off; off += (size_t)Nn * GATD;
    float* skip   = ws + off; off += (size_t)Nn * EMB;
    float* alpha  = ws + off; off += (size_t)E2 * HEADS;
    (void)ws_size; (void)n_in; (void)out_size;

    const int TB = 256;
    auto blocks  = [](long long t, int tb) { return (int)((t + tb - 1) / tb); };

    // 0) init accumulators
    init_ws_k<<<blocks((long long)Nn * 128, TB), TB, 0, stream>>>(
        la_sum, cnt, agg, denom, amax, outbuf, Nn);

    // 1) weighted in-degree + edge count
    degree_k<<<blocks(Ee, TB), TB, 0, stream>>>(dst, w, la_sum, cnt, Ee);

    // 2) h = x @ W1^T      (WMMA fp32)
    {
        long long waves = (long long)(Nn / 16) * (HID / 16);
        gemm_wmma_f32<HID><<<blocks(waves * 32, TB), TB, 0, stream>>>(
            x, W1, nullptr, h, Nn);
    }

    // 3) GCN neighbor scatter (self-loop handled analytically in finalize)
    gcn_scatter_k<<<blocks((long long)Ee * 32, TB), TB, 0, stream>>>(
        src, dst, w, la_sum, h, agg, Ee);

    // 4) hg = relu(agg + h/deg + b1)
    gcn_finalize_k<<<blocks((long long)Nn * HID, TB), TB, 0, stream>>>(
        agg, h, la_sum, b1, hg, Nn);

    // 5) xl = hg @ Wl^T + bl ; xr = hg @ Wr^T + br ; skip = x @ Ws^T + bs
    {
        long long waves = (long long)(Nn / 16) * (GATD / 16);
        gemm_wmma_f32<GATD><<<blocks(waves * 32, TB), TB, 0, stream>>>(
            hg, Wl, bl, xl, Nn);
        gemm_wmma_f32<GATD><<<blocks(waves * 32, TB), TB, 0, stream>>>(
            hg, Wr, br, xr, Nn);
        long long waves_s = (long long)(Nn / 16) * (EMB / 16);
        gemm_wmma_f32<EMB><<<blocks(waves_s * 32, TB), TB, 0, stream>>>(
            x, Ws, bs, skip, Nn);
    }

    // 6) GATv2 attention logits + per-target max
    gat_alpha_k<<<blocks((long long)E2 * 4, TB), TB, 0, stream>>>(
        src, dst, w, la_sum, cnt, xl, xr, We, att, alpha, amax, Ee, Nn);

    // 7) exp + per-target sum
    gat_expsum_k<<<blocks((long long)E2 * 4, TB), TB, 0, stream>>>(
        dst, alpha, amax, denom, Ee, Nn);

    // 8) attention-weighted scatter
    gat_scatter_k<<<blocks((long long)E2 * 32, TB), TB, 0, stream>>>(
        src, dst, xl, alpha, denom, outbuf, Ee, Nn);

    // 9) head mean + bias + gated skip
    final_out_k<<<blocks((long long)Nn * EMB, TB), TB, 0, stream>>>(
        outbuf, skip, bg, sg, out, Nn);
}